// DCNv3NCHW_34462817583481
// MI455X (gfx1250) — compile-verified
//
#include <hip/hip_runtime.h>
#include <cmath>

typedef __attribute__((ext_vector_type(2))) float v2f;
typedef __attribute__((ext_vector_type(8))) float v8f;

typedef __attribute__((address_space(1))) int as1_int;
typedef __attribute__((address_space(3))) int as3_int;

namespace dcn {
constexpr int Cc = 256;           // channels
constexpr int Gg = 16;            // groups
constexpr int Pp = 9;             // K*K taps
constexpr int Nn = 4, Hh = 64, Ww = 64;
constexpr int HW = Hh * Ww;       // 4096
constexpr int ROWS = Nn * HW;     // 16384
constexpr int OFFD = Gg * Pp * 2; // 288
constexpr int MSKD = Gg * Pp;     // 144
constexpr int KD   = 256;         // GEMM reduction depth (always C)
constexpr float LN_EPS = 1e-5f;
}

// ---------------------------------------------------------------------------
// Async global -> LDS copy (CDNA5 GLOBAL_LOAD_ASYNC_TO_LDS_B32, ASYNCcnt).
// ---------------------------------------------------------------------------
__device__ __forceinline__ void async_g2l_b32(const float* gptr, float* lptr)
{
#if defined(__has_builtin) && __has_builtin(__builtin_amdgcn_global_load_async_to_lds_b32)
    __builtin_amdgcn_global_load_async_to_lds_b32(
        (as1_int*)(void*)const_cast<float*>(gptr),
        (as3_int*)(void*)lptr,
        0, 0);
#else
    *lptr = *gptr;
#endif
}

// Wait until at most N async load instructions remain outstanding.
// Async loads complete in order, so N>0 implements a software pipeline.
template<int N>
__device__ __forceinline__ void wait_async_le()
{
    asm volatile("s_wait_asynccnt %0" :: "i"(N) : "memory");
}

// ---------------------------------------------------------------------------
// Kernel 1: depthwise 3x3 conv (NCHW, zero pad) + bias + LayerNorm(C) + GELU
// One block per pixel, one thread per channel (8 wave32 / block).
// ---------------------------------------------------------------------------
__global__ __launch_bounds__(256)
void dwln_kernel(const float* __restrict__ x,
                 const float* __restrict__ dw_w,
                 const float* __restrict__ dw_b,
                 const float* __restrict__ ln_g,
                 const float* __restrict__ ln_b,
                 float* __restrict__ x1out)
{
    using namespace dcn;
    const int p  = blockIdx.x;          // 0..16383
    const int c  = threadIdx.x;         // 0..255
    const int n  = p >> 12;
    const int hw = p & 4095;
    const int h  = hw >> 6;
    const int w  = hw & 63;

    const float* __restrict__ xc = x + ((size_t)(n * Cc + c)) * HW;
    const float* __restrict__ wc = dw_w + c * 9;

    float acc = 0.0f;
#pragma unroll
    for (int dy = 0; dy < 3; ++dy) {
        const int yy = h + dy - 1;
        if ((unsigned)yy >= 64u) continue;
#pragma unroll
        for (int dx = 0; dx < 3; ++dx) {
            const int xx = w + dx - 1;
            if ((unsigned)xx >= 64u) continue;
            acc += xc[yy * 64 + xx] * wc[dy * 3 + dx];
        }
    }
    acc += dw_b[c];

    // LayerNorm across the 256 channels of this block
    __shared__ float rs[256];
    __shared__ float rq[256];
    rs[c] = acc;
    rq[c] = acc * acc;
    __syncthreads();
#pragma unroll
    for (int s = 128; s > 0; s >>= 1) {
        if (c < s) { rs[c] += rs[c + s]; rq[c] += rq[c + s]; }
        __syncthreads();
    }
    const float mu  = rs[0] * (1.0f / 256.0f);
    const float var = rq[0] * (1.0f / 256.0f) - mu * mu;
    float v = (acc - mu) * rsqrtf(var + LN_EPS) * ln_g[c] + ln_b[c];
    // exact GELU
    v = 0.5f * v * (1.0f + erff(v * 0.70710678118654752440f));
    x1out[(size_t)p * Cc + c] = v;
}

// ---------------------------------------------------------------------------
// Kernel 2: WMMA fp32 GEMM  C[M,Nc] = A[M,256] * W[256,Nc] + bias
//   GATHER_A    : A(row,k) gathered from NCHW x (fuses NCHW->NHWC transpose)
//   SCATTER_OUT : D written NCHW                (fuses NHWC->NCHW transpose)
//   HAS2        : wave computes 2 N-tiles (block tile 64x32) vs 1 (64x16)
// 128 threads = 4 wave32. Double-buffered async global->LDS staging; B is
// staged TRANSPOSED so A and B fragments both load as contiguous b64 pairs.
// Row pad = 36 words: 36m mod 64 distinct and bank-disjoint (mod 4) from the
// +2-word addresses of lanes 16..31.
// ---------------------------------------------------------------------------
template<bool GATHER_A, bool SCATTER_OUT, bool HAS2>
__global__ __launch_bounds__(128)
void gemm_wmma_f32(const float* __restrict__ A,
                   const float* __restrict__ Wm,
                   const float* __restrict__ bias,
                   float* __restrict__ Cout,
                   int Nc, int colbase,
                   const float* __restrict__ Xsrc)
{
    using namespace dcn;
    constexpr int BN      = HAS2 ? 32 : 16;  // block tile N
    constexpr int B_ITERS = HAS2 ? 8 : 4;    // B elements per thread per stage
    constexpr int INFLIGHT = 16 + B_ITERS;   // async instrs per wave per stage

    __shared__ float As [2][64][36];   // As [buf][row][k]
    __shared__ float BsT[2][32][36];   // BsT[buf][col][k]  (transposed!)

    const int tid  = threadIdx.x;
    const int lane = tid & 31;
    const int wave = tid >> 5;
    const int row0 = blockIdx.x * 64;
    const int col0 = colbase + blockIdx.y * BN;
    const int m    = lane & 15;    // row (A) / col (B,C) within tile
    const int half = lane >> 4;    // K pair selector per ISA 16x4 f32 layout

    auto issue_stage = [&](int k0, int buf) {
        // A tile 64x32: 16 elements/thread
#pragma unroll
        for (int i = 0; i < 16; ++i) {
            const int idx = tid + i * 128;
            const int r   = idx >> 5;
            const int kc  = idx & 31;
            const float* src;
            if (GATHER_A) {
                const int row = row0 + r;
                const int n   = row >> 12;
                const int hw  = row & 4095;
                src = &Xsrc[((size_t)(n * Cc + (k0 + kc))) * HW + hw];
            } else {
                src = &A[(size_t)(row0 + r) * KD + (k0 + kc)];
            }
            async_g2l_b32(src, &As[buf][r][kc]);
        }
        // B tile 32xBN, stored transposed: BsT[col][k]
#pragma unroll
        for (int i = 0; i < B_ITERS; ++i) {
            const int idx = tid + i * 128;
            const int r   = HAS2 ? (idx >> 5) : (idx >> 4);
            const int cc  = HAS2 ? (idx & 31) : (idx & 15);
            async_g2l_b32(&Wm[(size_t)(k0 + r) * Nc + (col0 + cc)],
                          &BsT[buf][cc][r]);
        }
    };

    v8f acc0 = {};
    v8f acc1 = {};

    issue_stage(0, 0);
    for (int k0 = 0; k0 < KD; k0 += 32) {
        const int  buf  = (k0 >> 5) & 1;
        const bool more = (k0 + 32) < KD;
        if (more) {
            issue_stage(k0 + 32, buf ^ 1);
            if (k0 + 64 < KD) {
                __builtin_prefetch(&Wm[(size_t)(k0 + 64) * Nc + col0], 0, 3);
            }
            wait_async_le<INFLIGHT>();   // stage k landed; stage k+1 in flight
        } else {
            wait_async_le<0>();
        }
        __syncthreads();

#pragma unroll
        for (int kk = 0; kk < 32; kk += 4) {
            const int kb = kk + (half << 1);
            v2f a, b0;
            // A 16x4 f32: lanes 0-15 -> K = kk,kk+1 ; lanes 16-31 -> K = kk+2,kk+3
            a.x = As[buf][(wave << 4) + m][kb];
            a.y = As[buf][(wave << 4) + m][kb + 1];
            // B 4x16 f32: lane holds W[kb..kb+1][n=m] -> contiguous in BsT
            b0.x = BsT[buf][m][kb];
            b0.y = BsT[buf][m][kb + 1];
            acc0 = __builtin_amdgcn_wmma_f32_16x16x4_f32(
                false, a, false, b0, (short)0, acc0, false, false);
            if (HAS2) {
                v2f b1;
                b1.x = BsT[buf][16 + m][kb];
                b1.y = BsT[buf][16 + m][kb + 1];
                acc1 = __builtin_amdgcn_wmma_f32_16x16x4_f32(
                    false, a, false, b1, (short)0, acc1, false, false);
            }
        }
        __syncthreads();
    }

    const int colA = col0 + m;
    const int colB = colA + 16;
    const float bA = bias[colA];
    const float bB = HAS2 ? bias[colB] : 0.0f;
#pragma unroll
    for (int r = 0; r < 8; ++r) {
        // C 16x16 f32 layout: VGPR r, lanes 0-15 -> M=r, lanes 16-31 -> M=r+8
        const int row = row0 + (wave << 4) + r + (half << 3);
        const float v0 = acc0[r] + bA;
        const float v1 = acc1[r] + bB;
        if (SCATTER_OUT) {
            const int n  = row >> 12;
            const int hw = row & 4095;
            Cout[((size_t)(n * Cc + colA)) * HW + hw] = v0;
            if (HAS2) Cout[((size_t)(n * Cc + colB)) * HW + hw] = v1;
        } else {
            Cout[(size_t)row * Nc + colA] = v0;
            if (HAS2) Cout[(size_t)row * Nc + colB] = v1;
        }
    }
}

// ---------------------------------------------------------------------------
// Kernel 3: per-group softmax over 9 taps + bilinear deformable sampling.
// One block per pixel; thread t -> (g = t>>4, cc = t&15); 16 consecutive
// lanes fetch 16 consecutive channels -> coalesced 64B corner loads.
// ---------------------------------------------------------------------------
__global__ __launch_bounds__(256)
void sample_kernel(const float* __restrict__ xproj,
                   const float* __restrict__ offs,
                   const float* __restrict__ mlog,
                   float* __restrict__ samp)
{
    using namespace dcn;
    const int p  = blockIdx.x;
    const int t  = threadIdx.x;
    const int n  = p >> 12;
    const int hw = p & 4095;
    const int h  = hw >> 6;
    const int w  = hw & 63;

    __shared__ float soff[OFFD];   // 288
    __shared__ float sml[MSKD];    // 144
    for (int i = t; i < OFFD; i += 256) soff[i] = offs[(size_t)p * OFFD + i];
    if (t < MSKD) sml[t] = mlog[(size_t)p * MSKD + t];
    __syncthreads();

    const int g  = t >> 4;
    // softmax over the 9 taps of this group (done redundantly per lane; cheap)
    float mx = -1e30f;
#pragma unroll
    for (int pp = 0; pp < 9; ++pp) mx = fmaxf(mx, sml[g * 9 + pp]);
    float wts[9];
    float sume = 0.0f;
#pragma unroll
    for (int pp = 0; pp < 9; ++pp) {
        const float e = __expf(sml[g * 9 + pp] - mx);
        wts[pp] = e;
        sume += e;
    }
    const float inv = 1.0f / sume;

    const float dtab[3] = { -1.0f, 0.0f, 1.0f };
    const float* __restrict__ xpn = xproj + (size_t)n * HW * Cc;
    const int ch = t;              // g*16 + cc == t

    float acc = 0.0f;
#pragma unroll
    for (int pp = 0; pp < 9; ++pp) {
        const float ox = soff[(g * 9 + pp) * 2 + 0];
        const float oy = soff[(g * 9 + pp) * 2 + 1];
        // padded-map pixel coords; px = loc_x*Win - 0.5 simplifies to:
        const float px = (float)(w + 1) + dtab[pp / 3] + ox;
        const float py = (float)(h + 1) + dtab[pp % 3] + oy;
        const float fx = floorf(px), fy = floorf(py);
        const float wx = px - fx,  wy = py - fy;
        const int x0 = (int)fx, y0 = (int)fy;
        const float mw = wts[pp] * inv;

        float v00 = 0.f, v01 = 0.f, v10 = 0.f, v11 = 0.f;
        // padded map is zero outside [1,64]^2; interior -> xproj[(iy-1, ix-1)]
        if ((unsigned)(y0 - 1) < 64u && (unsigned)(x0 - 1) < 64u)
            v00 = xpn[((size_t)((y0 - 1) * 64 + (x0 - 1))) * Cc + ch];
        if ((unsigned)(y0 - 1) < 64u && (unsigned)(x0) < 64u)
            v01 = xpn[((size_t)((y0 - 1) * 64 + x0)) * Cc + ch];
        if ((unsigned)(y0) < 64u && (unsigned)(x0 - 1) < 64u)
            v10 = xpn[((size_t)(y0 * 64 + (x0 - 1))) * Cc + ch];
        if ((unsigned)(y0) < 64u && (unsigned)(x0) < 64u)
            v11 = xpn[((size_t)(y0 * 64 + x0)) * Cc + ch];

        acc += mw * ((1.f - wy) * ((1.f - wx) * v00 + wx * v01) +
                     wy         * ((1.f - wx) * v10 + wx * v11));
    }
    samp[(size_t)p * Cc + t] = acc;
}

// ---------------------------------------------------------------------------
extern "C" void kernel_launch(void* const* d_in, const int* in_sizes, int n_in,
                              void* d_out, int out_size, void* d_ws, size_t ws_size,
                              hipStream_t stream)
{
    using namespace dcn;
    const float* x      = (const float*)d_in[0];
    const float* dw_w   = (const float*)d_in[1];
    const float* dw_b   = (const float*)d_in[2];
    const float* ln_g   = (const float*)d_in[3];
    const float* ln_b   = (const float*)d_in[4];
    const float* off_w  = (const float*)d_in[5];
    const float* off_b  = (const float*)d_in[6];
    const float* mask_w = (const float*)d_in[7];
    const float* mask_b = (const float*)d_in[8];
    const float* inp_w  = (const float*)d_in[9];
    const float* inp_b  = (const float*)d_in[10];
    const float* out_w  = (const float*)d_in[11];
    const float* out_b  = (const float*)d_in[12];
    float* out = (float*)d_out;

    float* ws    = (float*)d_ws;
    float* x1    = ws;                           // 16384*256
    float* xproj = x1    + (size_t)ROWS * Cc;    // 16384*256
    float* offs  = xproj + (size_t)ROWS * Cc;    // 16384*288
    float* mlog  = offs  + (size_t)ROWS * OFFD;  // 16384*144
    float* samp  = mlog  + (size_t)ROWS * MSKD;  // 16384*256

    dwln_kernel<<<ROWS, 256, 0, stream>>>(x, dw_w, dw_b, ln_g, ln_b, x1);

    dim3 blk(128);
    // input_proj: A gathered straight from NCHW x (transpose fused). 256 cols.
    gemm_wmma_f32<true, false, true><<<dim3(ROWS / 64, Cc / 32), blk, 0, stream>>>(
        nullptr, inp_w, inp_b, xproj, Cc, 0, x);
    // offsets: 288 cols = 9 x 32
    gemm_wmma_f32<false, false, true><<<dim3(ROWS / 64, OFFD / 32), blk, 0, stream>>>(
        x1, off_w, off_b, offs, OFFD, 0, nullptr);
    // mask logits: 144 cols = 4 x 32 + 1 x 16 (softmax fused into sampling)
    gemm_wmma_f32<false, false, true><<<dim3(ROWS / 64, 4), blk, 0, stream>>>(
        x1, mask_w, mask_b, mlog, MSKD, 0, nullptr);
    gemm_wmma_f32<false, false, false><<<dim3(ROWS / 64, 1), blk, 0, stream>>>(
        x1, mask_w, mask_b, mlog, MSKD, 128, nullptr);

    sample_kernel<<<ROWS, 256, 0, stream>>>(xproj, offs, mlog, samp);

    // output_proj with fused NHWC -> NCHW scatter. 256 cols.
    gemm_wmma_f32<false, true, true><<<dim3(ROWS / 64, Cc / 32), blk, 0, stream>>>(
        samp, out_w, out_b, out, Cc, 0, nullptr);
}